// MyCostModule_40570261078651
// MI455X (gfx1250) — compile-verified
//
#include <hip/hip_runtime.h>

// Problem constants (match reference)
constexpr int B = 64, N = 512, R = 40, L = 20;
constexpr float MEAN_STOP = 60.0f;
constexpr int CHUNKS = 32;                 // blocks per batch in main kernel
constexpr int ELEMS_PER_BLOCK = (N * N) / CHUNKS;   // 8192
constexpr int ACC_STRIDE = 16;             // per-batch scalar accumulator slots in ws
// ws layout (floats): [0, B*16) scalar accs, [B*16, B*16 + B*R) riders
constexpr int WS_ACC = 0;
constexpr int WS_RIDERS = B * ACC_STRIDE;
constexpr int WS_FLOATS = B * ACC_STRIDE + B * R;   // 3584
// d_out layout (floats): cost[64] | riders[2560] | tat[256] | total_route_time[64]
constexpr int OUT_RIDERS = B;
constexpr int OUT_TAT = B + B * R;
constexpr int OUT_TRT = B + B * R + B * 4;

typedef float v2f __attribute__((ext_vector_type(2)));
typedef float v8f __attribute__((ext_vector_type(8)));

// Cross-lane (wave32) sum using the CDNA5 matrix pipe.
// A(16x4 f32) = per-lane value in K=0 (V0), 0 in K=1 (V1); B(4x16) = ones.
// D[m,n] = acc[m] + acc[m+16] for every column n. Each lane sums its 8 D
// VGPRs (rows 0..7 for lanes 0-15, rows 8..15 for lanes 16-31); lane 0 + lane 16
// then hold the two halves of the full 32-lane sum.
__device__ __forceinline__ float wave_sum_wmma(float v) {
    v2f a; a[0] = v;     a[1] = 0.0f;
    v2f o; o[0] = 1.0f;  o[1] = 1.0f;
    v8f c = {0.f, 0.f, 0.f, 0.f, 0.f, 0.f, 0.f, 0.f};
    v8f d = __builtin_amdgcn_wmma_f32_16x16x4_f32(false, a, false, o,
                                                  (short)0, c, false, false);
    float p = ((d[0] + d[1]) + (d[2] + d[3])) + ((d[4] + d[5]) + (d[6] + d[7]));
    return __shfl(p, 0, 32) + __shfl(p, 16, 32);
}

__global__ void init_ws_kernel(float* ws, int n) {
    int i = blockIdx.x * blockDim.x + threadIdx.x;
    if (i < n) ws[i] = 0.0f;
}

// Fused streaming reduction over the B x N x N arrays.
// grid = (CHUNKS, B), block = 256 threads (8 waves).
__global__ __launch_bounds__(256) void cost_reduce_kernel(
    const float* __restrict__ demand,
    const float* __restrict__ drive,
    const float* __restrict__ transit,
    const unsigned char* __restrict__ haspath,
    const int* __restrict__ ntrans,
    const int* __restrict__ rseq,
    float* __restrict__ acc,       // [B][ACC_STRIDE]
    float* __restrict__ riders)    // [B][R]
{
    __shared__ float ldsRiders[8][R];   // per-wave segment-sum copies
    const int tid  = threadIdx.x;
    const int wave = tid >> 5;
    const int b    = blockIdx.y;

    for (int i = tid; i < 8 * R; i += 256) ((float*)ldsRiders)[i] = 0.0f;
    __syncthreads();

    const size_t base = (size_t)b * N * N + (size_t)blockIdx.x * ELEMS_PER_BLOCK;

    float tat0 = 0.f, tat1 = 0.f, tat2 = 0.f, tat3 = 0.f;
    float dmdT = 0.f, uns = 0.f, tot = 0.f, ndis = 0.f, nedg = 0.f, mx = 0.f;

    const int groups = ELEMS_PER_BLOCK >> 2;   // 2048 groups of 4 elements
    for (int g = tid; g < groups; g += 256) {
        size_t e = base + ((size_t)g << 2);
        float4 dm = *(const float4*)(demand  + e);
        float4 dv = *(const float4*)(drive   + e);
        float4 tr = *(const float4*)(transit + e);
        int4   nt = *(const int4*)(ntrans    + e);
        unsigned hp4 = *(const unsigned*)(haspath + e);
        const int4* rs = (const int4*)(rseq + (e << 2));   // 4 route ids / element
#pragma unroll
        for (int j = 0; j < 4; ++j) {
            float d  = (&dm.x)[j];
            float t  = (&tr.x)[j];
            float dr = (&dv.x)[j];
            int   nv = (&nt.x)[j];
            bool  np = ((hp4 >> (8 * j)) & 0xffu) == 0u;   // no path
            int  nte = np ? 3 : nv;                        // effective transfers
            tat0 += (nte == 0) ? d : 0.0f;
            tat1 += (nte == 1) ? d : 0.0f;
            tat2 += (nte == 2) ? d : 0.0f;
            tat3 += (nte >= 3) ? d : 0.0f;
            dmdT += np ? 0.0f : d * t;
            uns  += np ? d : 0.0f;
            tot  += d;
            ndis += (np && d > 0.0f) ? 1.0f : 0.0f;
            nedg += (d > 0.0f) ? 1.0f : 0.0f;
            mx    = fmaxf(mx, dr);
            int4 rv = rs[j];
            if (rv.x >= 0) atomicAdd(&ldsRiders[wave][rv.x], d);
            if (rv.y >= 0) atomicAdd(&ldsRiders[wave][rv.y], d);
            if (rv.z >= 0) atomicAdd(&ldsRiders[wave][rv.z], d);
            if (rv.w >= 0) atomicAdd(&ldsRiders[wave][rv.w], d);
        }
    }
    __syncthreads();

    // Combine the 8 per-wave rider copies; one global atomic per route per block.
    for (int r = tid; r < R; r += 256) {
        float s = 0.f;
#pragma unroll
        for (int w = 0; w < 8; ++w) s += ldsRiders[w][r];
        atomicAdd(&riders[b * R + r], s);
    }

    // Wave-level scalar reductions via the matrix pipe (EXEC is all-ones here).
    float s0 = wave_sum_wmma(tat0);
    float s1 = wave_sum_wmma(tat1);
    float s2 = wave_sum_wmma(tat2);
    float s3 = wave_sum_wmma(tat3);
    float s4 = wave_sum_wmma(dmdT);
    float s5 = wave_sum_wmma(uns);
    float s6 = wave_sum_wmma(tot);
    float s7 = wave_sum_wmma(ndis);
    float s8 = wave_sum_wmma(nedg);
#pragma unroll
    for (int m = 16; m >= 1; m >>= 1) mx = fmaxf(mx, __shfl_xor(mx, m, 32));

    if ((tid & 31) == 0) {
        float* a = acc + b * ACC_STRIDE;
        atomicAdd(&a[0], s0); atomicAdd(&a[1], s1);
        atomicAdd(&a[2], s2); atomicAdd(&a[3], s3);
        atomicAdd(&a[4], s4); atomicAdd(&a[5], s5);
        atomicAdd(&a[6], s6); atomicAdd(&a[7], s7);
        atomicAdd(&a[8], s8);
        // drive_times >= 0 so uint bit-pattern ordering == float ordering
        atomicMax((unsigned int*)&a[9], __float_as_uint(mx));
    }
}

// Per-(batch, route) leg-time walk + route-length stats.
__global__ void route_legs_kernel(const float* __restrict__ drive,
                                  const int* __restrict__ routes,
                                  float* __restrict__ acc)
{
    int idx = blockIdx.x * blockDim.x + threadIdx.x;
    if (idx >= B * R) return;
    int b = idx / R, r = idx % R;
    const int* rt = routes + (size_t)(b * R + r) * L;
    int len = 0;
    float t = 0.f;
    int cur = rt[0];
#pragma unroll
    for (int i = 0; i < L - 1; ++i) {
        int nxt = rt[i + 1];
        len += (cur > -1);
        if (cur >= 0 && nxt >= 0) {
            size_t bo = (size_t)b * N * N;
            t += drive[bo + (size_t)cur * N + nxt]
               + drive[bo + (size_t)nxt * N + cur]
               + 2.0f * MEAN_STOP;
        }
        cur = nxt;
    }
    len += (cur > -1);
    float* a = acc + b * ACC_STRIDE;
    atomicAdd(&a[10], t);                                            // total_route_time
    float delta = (len == 0) ? 0.f : fmaxf((float)(2 - len), 0.f);
    atomicAdd(&a[11], delta);                                        // n_stops_oob
    atomicAdd(&a[12], (len > 0) ? 1.0f : 0.0f);                      // n_routes_used
}

__global__ void finalize_kernel(const float* __restrict__ acc,
                                const float* __restrict__ riders,
                                float* __restrict__ out)
{
    int b = blockIdx.x, t = threadIdx.x;
    const float* a = acc + b * ACC_STRIDE;
    if (t < R)  out[OUT_RIDERS + b * R + t] = riders[b * R + t];
    if (t < 4)  out[OUT_TAT + b * 4 + t]    = a[t];
    if (t == 0) {
        float time_norm = a[9];                     // bits written by atomicMax, valid f32
        float served    = a[6] - a[5];
        float mean_dmd  = a[4] / (served + 1e-6f);
        float norm_dmd  = mean_dmd / time_norm;
        float nru       = a[12];
        float norm_rt   = a[10] / (time_norm * nru + 1e-6f);
        float frac_unc  = a[7] / a[8];
        float frac_oob  = a[11] / (nru * 2.0f + 1e-6f);
        out[b]           = 0.5f * norm_dmd + 0.5f * norm_rt + 5.0f * (frac_unc + frac_oob);
        out[OUT_TRT + b] = a[10];
    }
}

extern "C" void kernel_launch(void* const* d_in, const int* in_sizes, int n_in,
                              void* d_out, int out_size, void* d_ws, size_t ws_size,
                              hipStream_t stream) {
    (void)in_sizes; (void)n_in; (void)out_size; (void)ws_size;
    const float*         demand  = (const float*)d_in[0];
    const float*         drive   = (const float*)d_in[1];
    const float*         transit = (const float*)d_in[2];
    const unsigned char* haspath = (const unsigned char*)d_in[3];
    const int*           ntrans  = (const int*)d_in[4];
    const int*           routes  = (const int*)d_in[5];
    const int*           rseq    = (const int*)d_in[6];
    float* ws  = (float*)d_ws;
    float* out = (float*)d_out;
    float* acc    = ws + WS_ACC;
    float* riders = ws + WS_RIDERS;

    init_ws_kernel<<<(WS_FLOATS + 255) / 256, 256, 0, stream>>>(ws, WS_FLOATS);
    cost_reduce_kernel<<<dim3(CHUNKS, B), 256, 0, stream>>>(
        demand, drive, transit, haspath, ntrans, rseq, acc, riders);
    route_legs_kernel<<<(B * R + 255) / 256, 256, 0, stream>>>(drive, routes, acc);
    finalize_kernel<<<B, 64, 0, stream>>>(acc, riders, out);
}